// DressedQuantumNet2_35983236006318
// MI455X (gfx1250) — compile-verified
//
#include <hip/hip_runtime.h>
#include <math.h>

// fp32 WMMA fragments (wave32): A 16x4 -> 2 VGPR/lane, C/D 16x16 -> 8 VGPR/lane
typedef __attribute__((ext_vector_type(2))) float v2f;
typedef __attribute__((ext_vector_type(8))) float v8f;

#define FDIM 512
#define ROWS_PER_WAVE 32
#define WAVES_PER_BLOCK 8
#define ROWS_PER_BLOCK (ROWS_PER_WAVE * WAVES_PER_BLOCK) // 256

// ---- fully-unrolled 4-qubit gate helpers (wire w -> mask 1<<(3-w)) ----

template <int CM, int TM>
__device__ __forceinline__ void cnot_g(float sr[16], float si[16]) {
#pragma unroll
  for (int i = 0; i < 16; ++i) {
    if ((i & CM) && !(i & TM)) {
      const int j = i | TM;
      float t;
      t = sr[i]; sr[i] = sr[j]; sr[j] = t;
      t = si[i]; si[i] = si[j]; si[j] = t;
    }
  }
}

template <int M>
__device__ __forceinline__ void ry_g(float sr[16], float si[16], float c, float s) {
#pragma unroll
  for (int i = 0; i < 16; ++i) {
    if (!(i & M)) {
      const int j = i | M;
      const float r0 = sr[i], i0 = si[i], r1 = sr[j], i1 = si[j];
      sr[i] = c * r0 - s * r1;  si[i] = c * i0 - s * i1;
      sr[j] = s * r0 + c * r1;  si[j] = s * i0 + c * i1;
    }
  }
}

template <int M>
__device__ __forceinline__ void rz_g(float sr[16], float si[16], float ct, float st) {
  // bit==0: multiply by e^{-i t} = (ct, -st); bit==1: e^{+i t} = (ct, +st)
#pragma unroll
  for (int i = 0; i < 16; ++i) {
    const float sg = (i & M) ? st : -st;
    const float r = sr[i], im = si[i];
    sr[i] = r * ct - im * sg;
    si[i] = r * sg + im * ct;
  }
}

__device__ __forceinline__ void entangle_g(float sr[16], float si[16]) {
  cnot_g<8, 4>(sr, si);  // CNOT(0,1)
  cnot_g<2, 1>(sr, si);  // CNOT(2,3)
  cnot_g<4, 2>(sr, si);  // CNOT(1,2)
}

#define RY_LAYER(kk)                                    \
  do {                                                  \
    ry_g<8>(sr, si, cq[(kk)*4 + 0], sq[(kk)*4 + 0]);    \
    ry_g<4>(sr, si, cq[(kk)*4 + 1], sq[(kk)*4 + 1]);    \
    ry_g<2>(sr, si, cq[(kk)*4 + 2], sq[(kk)*4 + 2]);    \
    ry_g<1>(sr, si, cq[(kk)*4 + 3], sq[(kk)*4 + 3]);    \
  } while (0)

#define RZ_LAYER(kk)                                    \
  do {                                                  \
    rz_g<8>(sr, si, cq[(kk)*4 + 0], sq[(kk)*4 + 0]);    \
    rz_g<4>(sr, si, cq[(kk)*4 + 1], sq[(kk)*4 + 1]);    \
    rz_g<2>(sr, si, cq[(kk)*4 + 2], sq[(kk)*4 + 2]);    \
    rz_g<1>(sr, si, cq[(kk)*4 + 3], sq[(kk)*4 + 3]);    \
  } while (0)

__global__ __launch_bounds__(256) void dqnet_fused_kernel(
    const float* __restrict__ x,       // [B,512]
    const float* __restrict__ pre_w,   // [4,512]
    const float* __restrict__ pre_b,   // [4]
    const float* __restrict__ qw,      // [6,4]
    const float* __restrict__ post_w,  // [2,4]
    const float* __restrict__ post_b,  // [2]
    float* __restrict__ out)           // [B,2]
{
  __shared__ float s_w[4 * FDIM];                          // pre_w staged, 8KB
  __shared__ float s_ang[WAVES_PER_BLOCK][ROWS_PER_WAVE][4]; // angle shuffle, 4KB

  const int tid  = threadIdx.x;
  const int wave = tid >> 5;
  const int lane = tid & 31;

  for (int i = tid; i < 4 * FDIM; i += 256) s_w[i] = pre_w[i];
  __syncthreads();

  const int rowBase = blockIdx.x * ROWS_PER_BLOCK + wave * ROWS_PER_WAVE;
  const int mrow  = lane & 15;            // M (tile row) / N (tile col) index
  const int khalf = (lane >> 4) << 1;     // lanes 16-31 carry K+2,K+3

  // A: x rows (two 16-row tiles per wave).  B: pre_w^T padded to 16 cols.
  const float* a0p = x + (size_t)(rowBase + mrow) * FDIM + khalf;
  const float* a1p = x + (size_t)(rowBase + 16 + mrow) * FDIM + khalf;
  const float  bmask = (mrow < 4) ? 1.0f : 0.0f;           // cols >=4 are zero
  const float* bp = s_w + (mrow & 3) * FDIM + khalf;

  v8f c0 = {};
  v8f c1 = {};
#pragma unroll 4
  for (int k = 0; k < FDIM; k += 4) {
    v2f a0, a1, b;
    a0[0] = a0p[k]; a0[1] = a0p[k + 1];
    a1[0] = a1p[k]; a1[1] = a1p[k + 1];
    b[0] = bp[k] * bmask;
    b[1] = bp[k + 1] * bmask;
    c0 = __builtin_amdgcn_wmma_f32_16x16x4_f32(false, a0, false, b, (short)0, c0, false, false);
    c1 = __builtin_amdgcn_wmma_f32_16x16x4_f32(false, a1, false, b, (short)0, c1, false, false);
  }

  // Scatter the 4 valid angle columns: C/D layout is VGPR r, lanes 0-15 -> (M=r, N=lane),
  // lanes 16-31 -> (M=8+r, N=lane-16).
  if (mrow < 4) {
    const int half = lane >> 4;
#pragma unroll
    for (int r = 0; r < 8; ++r) {
      const int m = half * 8 + r;
      s_ang[wave][m][mrow]      = c0[r];
      s_ang[wave][16 + m][mrow] = c1[r];
    }
  }
  __syncthreads();

  // ---- per-lane 4-qubit circuit (16 complex amplitudes in registers) ----
  float a[4];
#pragma unroll
  for (int w = 0; w < 4; ++w) a[w] = s_ang[wave][lane][w] + pre_b[w];

  float cq[24], sq[24];
#pragma unroll
  for (int i = 0; i < 24; ++i) {
    const float t = 0.5f * qw[i];
    cq[i] = __cosf(t);
    sq[i] = __sinf(t);
  }

  // front layer: H then RY(angle) on each wire -> real product state
  float sr[16], si[16];
  {
    const float inv = 0.70710678118654752f;
    float lo[4], hi[4];
#pragma unroll
    for (int w = 0; w < 4; ++w) {
      const float t  = 0.5f * a[w];
      const float ct = __cosf(t), st = __sinf(t);
      lo[w] = (ct - st) * inv;
      hi[w] = (ct + st) * inv;
    }
#pragma unroll
    for (int i = 0; i < 16; ++i) {
      float p = (i & 8) ? hi[0] : lo[0];
      p *= (i & 4) ? hi[1] : lo[1];
      p *= (i & 2) ? hi[2] : lo[2];
      p *= (i & 1) ? hi[3] : lo[3];
      sr[i] = p;
      si[i] = 0.0f;
    }
  }

  // var layers pass 1: entangle + RY(qw[k]) + RZ(qw[k])
  entangle_g(sr, si); RY_LAYER(0); RZ_LAYER(0);
  entangle_g(sr, si); RY_LAYER(1); RZ_LAYER(1);
  entangle_g(sr, si); RY_LAYER(2); RZ_LAYER(2);
  // var layers pass 2: entangle + RY(qw[k]) + RZ(qw[3+k])
  entangle_g(sr, si); RY_LAYER(0); RZ_LAYER(3);
  entangle_g(sr, si); RY_LAYER(1); RZ_LAYER(4);
  entangle_g(sr, si); RY_LAYER(2); RZ_LAYER(5);
  // final entanglement
  entangle_g(sr, si);

  // <Z_w> expectations + post layer
  float e0 = 0.f, e1 = 0.f, e2 = 0.f, e3 = 0.f;
#pragma unroll
  for (int i = 0; i < 16; ++i) {
    const float p = sr[i] * sr[i] + si[i] * si[i];
    e0 += (i & 8) ? -p : p;
    e1 += (i & 4) ? -p : p;
    e2 += (i & 2) ? -p : p;
    e3 += (i & 1) ? -p : p;
  }
  const float o0 = post_b[0] + post_w[0] * e0 + post_w[1] * e1 + post_w[2] * e2 + post_w[3] * e3;
  const float o1 = post_b[1] + post_w[4] * e0 + post_w[5] * e1 + post_w[6] * e2 + post_w[7] * e3;

  const int row = rowBase + lane;
  reinterpret_cast<float2*>(out)[row] = make_float2(o0, o1);
}

extern "C" void kernel_launch(void* const* d_in, const int* in_sizes, int n_in,
                              void* d_out, int out_size, void* d_ws, size_t ws_size,
                              hipStream_t stream) {
  const float* x      = (const float*)d_in[0];
  const float* pre_w  = (const float*)d_in[1];
  const float* pre_b  = (const float*)d_in[2];
  const float* qw     = (const float*)d_in[3];
  const float* post_w = (const float*)d_in[4];
  const float* post_b = (const float*)d_in[5];
  float* out = (float*)d_out;

  const int B = in_sizes[0] / FDIM;          // 65536
  dim3 grid(B / ROWS_PER_BLOCK);             // 256 blocks
  dim3 block(256);                           // 8 waves
  hipLaunchKernelGGL(dqnet_fused_kernel, grid, block, 0, stream,
                     x, pre_w, pre_b, qw, post_w, post_b, out);
}